// EncLoss_52725018526410
// MI455X (gfx1250) — compile-verified
//
#include <hip/hip_runtime.h>
#include <hip/hip_bf16.h>
#include <stdint.h>

// Problem geometry (fixed by the reference).
#define NCLS 19
#define BATCH 16
#define HH 512
#define WW 512
#define HW (HH * WW)                  // 262144 pixels per channel plane
#define TILE 1024                     // pixels handled per block
#define HALF 512                      // pixels per TDM descriptor (double buffer)
#define NBLOCKS (BATCH * HW / TILE)   // 4096
#define NTHREADS 256                  // 8 waves; 2 pixels/thread/half

typedef unsigned int u32x4 __attribute__((ext_vector_type(4)));
typedef int          i32x8 __attribute__((ext_vector_type(8)));
typedef int          i32x4 __attribute__((ext_vector_type(4)));

// ---------------------------------------------------------------------------
// Issue one TDM descriptor: 2D tile (HALF pixels x NCLS channels, row stride
// HW elements) from global -> LDS. Completion tracked by TENSORcnt.
// ---------------------------------------------------------------------------
__device__ __forceinline__ void tdm_load_tile(const float* gptr, const float* lptr) {
    const uint64_t gaddr = (uint64_t)(uintptr_t)gptr;
    // Generic pointers to LDS carry the byte offset in the low 32 bits.
    const unsigned lds = (unsigned)(uintptr_t)lptr;

    // D# group 0: [1:0]=count=1, [63:32]=lds_addr,
    //             [120:64]=global_addr, [127:126]=type=2
    u32x4 g0;
    g0.x = 1u;
    g0.y = lds;
    g0.z = (unsigned)(gaddr & 0xffffffffu);
    g0.w = (unsigned)((gaddr >> 32) & 0x01ffffffu) | (2u << 30);

    // D# group 1:
    //  [17:16]   data_size = 2 (4-byte elements)
    //  [79:48]   tensor_dim0 = HW
    //  [111:80]  tensor_dim1 = NCLS
    //  [127:112] tile_dim0 = HALF
    //  [143:128] tile_dim1 = NCLS   (tile_dim2 = 0)
    //  [207:160] tensor_dim0_stride = HW (elements between channel rows)
    i32x8 g1;
    g1[0] = (int)(2u << 16);
    g1[1] = (int)((unsigned)(HW & 0xffff) << 16);
    g1[2] = (int)(((unsigned)HW >> 16) & 0xffffu) | (NCLS << 16);
    g1[3] = (int)((unsigned)HALF << 16);
    g1[4] = NCLS;
    g1[5] = (int)HW;
    g1[6] = 0;
    g1[7] = 0;

    i32x4 gz4 = {0, 0, 0, 0};             // groups 2/3 unused (2D tensor)
    i32x8 gz8 = {0, 0, 0, 0, 0, 0, 0, 0};
    __builtin_amdgcn_tensor_load_to_lds(g0, g1, gz4, gz4, gz8, 0);
}

// ---------------------------------------------------------------------------
// Per-half compute: 2 pixels per thread from the staged LDS tile.
// ---------------------------------------------------------------------------
__device__ __forceinline__ void process_half(const float* __restrict__ buf,
                                             const int t0, const int t1,
                                             const int tid,
                                             float& nllsum, unsigned& mask) {
    float a[NCLS], b[NCLS];
#pragma unroll
    for (int c = 0; c < NCLS; ++c) {          // ds_load_b64, conflict-free
        const float2 p = *(const float2*)&buf[c * HALF + tid * 2];
        a[c] = p.x;
        b[c] = p.y;
    }
    float m0 = a[0], m1 = b[0];
#pragma unroll
    for (int c = 1; c < NCLS; ++c) {
        m0 = fmaxf(m0, a[c]);
        m1 = fmaxf(m1, b[c]);
    }
    float s0 = 0.0f, s1 = 0.0f;
#pragma unroll
    for (int c = 0; c < NCLS; ++c) {
        s0 += __expf(a[c] - m0);
        s1 += __expf(b[c] - m1);
    }
    // Dynamic-index gather straight from LDS (bank-conflict-free: HALF%64==0)
    const float vt0 = buf[t0 * HALF + tid * 2 + 0];
    const float vt1 = buf[t1 * HALF + tid * 2 + 1];
    nllsum += (__logf(s0) + m0 - vt0) + (__logf(s1) + m1 - vt1);
    mask   |= (1u << t0) | (1u << t1);
}

// ---------------------------------------------------------------------------
// Zero the per-batch class-presence masks (graph-safe re-init every launch).
// ---------------------------------------------------------------------------
__global__ void EncLoss_zero_pres(unsigned* pres) {
    if (threadIdx.x < BATCH) pres[threadIdx.x] = 0u;
}

// ---------------------------------------------------------------------------
// Main kernel: double-buffered TDM pipeline.
//   issue DMA(half0); issue DMA(half1); wait<=1; compute half0 (DMA half1 in
//   flight); wait<=0; compute half1; block-reduce.
// ---------------------------------------------------------------------------
__global__ __launch_bounds__(NTHREADS) void EncLoss_ce_main(
        const float* __restrict__ out0,
        const int*   __restrict__ target,
        float*       __restrict__ partial,
        unsigned*    __restrict__ pres) {
    __shared__ float    buf0[NCLS * HALF];    // 38 KB
    __shared__ float    buf1[NCLS * HALF];    // 38 KB
    __shared__ float    redf[NTHREADS];
    __shared__ unsigned redm[NTHREADS];

    const int bid = blockIdx.x;
    const int tid = threadIdx.x;
    const int b   = bid >> 8;                 // 256 tiles per batch image
    const int hw0 = (bid & 255) * TILE;
    const float* gbase = out0 + (size_t)b * NCLS * HW + hw0;

    // ---- Tensor Data Mover: issue both halves, wait for the first ----
    if (tid < 32) {                           // wave 0 drives the DMA engine
        tdm_load_tile(gbase,        buf0);
        tdm_load_tile(gbase + HALF, buf1);
        __builtin_amdgcn_s_wait_tensorcnt(1); // half0 landed (in-order)
    }
    __syncthreads();

    // Targets for this thread's 4 pixels (coalesced b64 loads).
    const int* tgt = target + (size_t)b * HW + hw0;
    const int2 ta = *(const int2*)&tgt[tid * 2];          // half 0
    const int2 tb = *(const int2*)&tgt[HALF + tid * 2];   // half 1

    float    nllsum = 0.0f;
    unsigned mask   = 0u;

    process_half(buf0, ta.x, ta.y, tid, nllsum, mask);    // overlaps DMA #2

    if (tid < 32) __builtin_amdgcn_s_wait_tensorcnt(0);   // half1 landed
    __syncthreads();

    process_half(buf1, tb.x, tb.y, tid, nllsum, mask);

    // ---- deterministic block reduction: NLL sum + presence mask ----
    redf[tid] = nllsum;
    redm[tid] = mask;
    __syncthreads();
#pragma unroll
    for (int off = NTHREADS / 2; off > 0; off >>= 1) {
        if (tid < off) {
            redf[tid] += redf[tid + off];
            redm[tid] |= redm[tid + off];
        }
        __syncthreads();
    }
    if (tid == 0) {
        partial[bid] = redf[0];
        atomicOr(&pres[b], redm[0]);          // bitwise OR: order-independent
    }
}

// ---------------------------------------------------------------------------
// Finalize: reduce 4096 partials (fixed order) + stable BCE-with-logits.
// ---------------------------------------------------------------------------
__global__ __launch_bounds__(NTHREADS) void EncLoss_finalize(
        const float*    __restrict__ partial,
        const unsigned* __restrict__ pres,
        const float*    __restrict__ out1,
        float*          __restrict__ out) {
    __shared__ float red[NTHREADS];
    const int tid = threadIdx.x;

    float s1 = 0.0f;
    for (int i = tid; i < NBLOCKS; i += NTHREADS) s1 += partial[i];

    float s2 = 0.0f;
    for (int i = tid; i < BATCH * NCLS; i += NTHREADS) {
        const float x  = out1[i];
        const float tt = (float)((pres[i / NCLS] >> (i % NCLS)) & 1u);
        // softplus(x) - t*x, numerically stable
        const float sp = fmaxf(x, 0.0f) + __logf(1.0f + __expf(-fabsf(x)));
        s2 += sp - tt * x;
    }

    red[tid] = s1 * (1.0f / (float)((size_t)BATCH * HW)) +
               s2 * (1.0f / (float)(BATCH * NCLS));
    __syncthreads();
#pragma unroll
    for (int off = NTHREADS / 2; off > 0; off >>= 1) {
        if (tid < off) red[tid] += red[tid + off];
        __syncthreads();
    }
    if (tid == 0) out[0] = red[0];
}

// ---------------------------------------------------------------------------
extern "C" void kernel_launch(void* const* d_in, const int* in_sizes, int n_in,
                              void* d_out, int out_size, void* d_ws, size_t ws_size,
                              hipStream_t stream) {
    const float* out0   = (const float*)d_in[0];   // [16,19,512,512] f32
    const float* out1   = (const float*)d_in[1];   // [16,19] f32
    const int*   target = (const int*)d_in[2];     // [16,512,512] int

    float*    partial = (float*)d_ws;                                   // 16 KB
    unsigned* pres    = (unsigned*)((char*)d_ws + NBLOCKS * sizeof(float));

    EncLoss_zero_pres<<<1, 64, 0, stream>>>(pres);
    EncLoss_ce_main<<<NBLOCKS, NTHREADS, 0, stream>>>(out0, target, partial, pres);
    EncLoss_finalize<<<1, NTHREADS, 0, stream>>>(partial, pres, out1, (float*)d_out);
}